// GCN_26087631356715
// MI455X (gfx1250) — compile-verified
//
#include <hip/hip_runtime.h>
#include <hip/hip_bf16.h>

// ---------------------------------------------------------------------------
// GCN (2x GraphConv + linear head) for MI455X / gfx1250.
//   - degree counts via global f32 atomics
//   - gather/scatter-add over edges (float4 gathers, f32 atomic adds; all
//     node buffers are L2-resident: 25.6 MB << 192 MB L2)
//   - node transform (agg*in_isqrt) @ W(64x64) + b via V_WMMA_F32_16X16X4_F32
//     (exact fp32 path), A staged through padded LDS, epilogue fuses bias,
//     LeakyReLU and next-layer out_isqrt scaling.
// ---------------------------------------------------------------------------

typedef __attribute__((ext_vector_type(2))) float v2f;
typedef __attribute__((ext_vector_type(8))) float v8f;

#define D_FEAT 64
#define LDS_STRIDE 68   // 64 + 4 pad: breaks bank conflicts, keeps 8B align

// ---- zero a float buffer ---------------------------------------------------
__global__ void gcn_zero_f32(float* __restrict__ p, int n) {
    int i = blockIdx.x * blockDim.x + threadIdx.x;
    if (i < n) p[i] = 0.0f;
}

// ---- degree counting: deg_out[src[e]] += 1 ; deg_in[dst[e]] += 1 -----------
__global__ void gcn_degree(const int* __restrict__ src, const int* __restrict__ dst,
                           float* __restrict__ deg_out, float* __restrict__ deg_in,
                           int nEdges) {
    int e = blockIdx.x * blockDim.x + threadIdx.x;
    if (e >= nEdges) return;
    atomicAdd(&deg_out[src[e]], 1.0f);
    atomicAdd(&deg_in[dst[e]], 1.0f);
}

// ---- d -> rsqrt(max(d,1)) in place ----------------------------------------
__global__ void gcn_deg_isqrt(float* __restrict__ d, int n) {
    int i = blockIdx.x * blockDim.x + threadIdx.x;
    if (i < n) d[i] = rsqrtf(fmaxf(d[i], 1.0f));
}

// ---- out[row,:] = x[row,:] * s[row]  (float4 granularity) ------------------
__global__ void gcn_scale_rows(const float* __restrict__ x, const float* __restrict__ s,
                               float* __restrict__ out, int nQuads /* = N*16 */) {
    int t = blockIdx.x * blockDim.x + threadIdx.x;
    if (t >= nQuads) return;
    int row = t >> 4;
    float sc = s[row];
    float4 v = ((const float4*)x)[t];
    v.x *= sc; v.y *= sc; v.z *= sc; v.w *= sc;
    ((float4*)out)[t] = v;
}

// ---- scatter-add: agg[dst[e],:] += h[src[e],:]  (16 lanes per edge) --------
__global__ void gcn_scatter(const float* __restrict__ h,
                            const int* __restrict__ src, const int* __restrict__ dst,
                            float* __restrict__ agg, int nEdges) {
    int t = blockIdx.x * blockDim.x + threadIdx.x;
    int e = t >> 4;
    if (e >= nEdges) return;
    int f = (t & 15) << 2;                 // feature quad 0..60
    int s = src[e], d = dst[e];
    float4 v = *(const float4*)(h + (size_t)s * D_FEAT + f);
    float* p = agg + (size_t)d * D_FEAT + f;
    atomicAdd(p + 0, v.x);
    atomicAdd(p + 1, v.y);
    atomicAdd(p + 2, v.z);
    atomicAdd(p + 3, v.w);
}

// ---- WMMA transform:  out = act((agg*in_isqrt) @ W + b) [* post_scale] -----
// Block = 128 threads = 4 waves; one 16-row tile per block, wave w owns
// columns [16w, 16w+16).  K=64 done as 16 steps of V_WMMA_F32_16X16X4_F32.
__global__ void __launch_bounds__(128)
gcn_transform_wmma(const float* __restrict__ agg,
                   const float* __restrict__ W,          // [64][64] row-major
                   const float* __restrict__ bias,       // [64]
                   const float* __restrict__ in_isqrt,   // [N]
                   const float* __restrict__ post_scale, // [N] or nullptr
                   float* __restrict__ out, int nNodes) {
    __shared__ float lds[16 * LDS_STRIDE];

    const int r0   = blockIdx.x * 16;
    const int tid  = threadIdx.x;
    const int wave = tid >> 5;
    const int lane = tid & 31;

    // Stage A tile (16 rows x 64 cols), fused with in_isqrt row scaling.
    {
        int idx = tid * 8;                 // 1024 floats / 128 threads
        int row = idx >> 6;
        int col = idx & 63;
        int grow = r0 + row; if (grow >= nNodes) grow = nNodes - 1;
        float s = in_isqrt[grow];
        const float* g = agg + (size_t)grow * D_FEAT + col;
        float* l = &lds[row * LDS_STRIDE + col];
        #pragma unroll
        for (int j = 0; j < 8; ++j) l[j] = g[j] * s;
    }
    __syncthreads();

    const int colbase = wave << 4;
    const int n     = lane & 15;
    const int khalf = lane >> 4;           // 0: K even pair, 1: K odd pair

    v8f acc = {};
    #pragma unroll
    for (int kk = 0; kk < 16; ++kk) {
        const int k0 = (kk << 2) + (khalf << 1);
        // A fragment: lane holds A[m=n][k0], A[m=n][k0+1]
        v2f a = *(const v2f*)&lds[n * LDS_STRIDE + k0];
        // B fragment: lane holds W[k0][colbase+n], W[k0+1][colbase+n]
        v2f b;
        b.x = W[(k0 + 0) * D_FEAT + colbase + n];
        b.y = W[(k0 + 1) * D_FEAT + colbase + n];
        acc = __builtin_amdgcn_wmma_f32_16x16x4_f32(
            /*neg_a=*/false, a, /*neg_b=*/false, b,
            /*c_mod=*/(short)0, acc, /*reuse_a=*/false, /*reuse_b=*/false);
    }

    // Epilogue: bias + LeakyReLU (+ optional out-degree pre-scale), store.
    // D layout: VGPR r, lanes 0-15 -> row r, col=lane; lanes 16-31 -> row 8+r.
    const int col = colbase + n;
    const float bv = bias[col];
    #pragma unroll
    for (int r = 0; r < 8; ++r) {
        int row = r0 + (khalf << 3) + r;
        if (row < nNodes) {
            float v = acc[r] + bv;
            v = (v > 0.0f) ? v : 0.01f * v;
            if (post_scale) v *= post_scale[row];
            out[(size_t)row * D_FEAT + col] = v;
        }
    }
}

// ---- classifier head: out[i,0:2] = h[i,:] @ Wc + bc ------------------------
__global__ void gcn_classify(const float* __restrict__ h,
                             const float* __restrict__ Wc,   // [64][2]
                             const float* __restrict__ bc,   // [2]
                             float* __restrict__ out, int nNodes) {
    int i = blockIdx.x * blockDim.x + threadIdx.x;
    if (i >= nNodes) return;
    float a0 = bc[0], a1 = bc[1];
    const float4* row = (const float4*)(h + (size_t)i * D_FEAT);
    #pragma unroll
    for (int q = 0; q < 16; ++q) {
        float4 v = row[q];
        int k = q << 2;
        a0 += v.x * Wc[(k+0)*2] + v.y * Wc[(k+1)*2] + v.z * Wc[(k+2)*2] + v.w * Wc[(k+3)*2];
        a1 += v.x * Wc[(k+0)*2+1] + v.y * Wc[(k+1)*2+1] + v.z * Wc[(k+2)*2+1] + v.w * Wc[(k+3)*2+1];
    }
    out[(size_t)i * 2 + 0] = a0;
    out[(size_t)i * 2 + 1] = a1;
}

// ---------------------------------------------------------------------------
extern "C" void kernel_launch(void* const* d_in, const int* in_sizes, int n_in,
                              void* d_out, int out_size, void* d_ws, size_t ws_size,
                              hipStream_t stream) {
    const float* x   = (const float*)d_in[0];
    const int*   src = (const int*)  d_in[1];
    const int*   dst = (const int*)  d_in[2];
    const float* W1  = (const float*)d_in[3];
    const float* b1  = (const float*)d_in[4];
    const float* W2  = (const float*)d_in[5];
    const float* b2  = (const float*)d_in[6];
    const float* Wc  = (const float*)d_in[7];
    const float* bc  = (const float*)d_in[8];
    float* out = (float*)d_out;

    const int N = in_sizes[0] / D_FEAT;    // 100000
    const int E = in_sizes[1];             // 1000000

    // workspace layout (floats)
    float* bufA      = (float*)d_ws;                 // N*64
    float* bufB      = bufA + (size_t)N * D_FEAT;    // N*64
    float* out_isqrt = bufB + (size_t)N * D_FEAT;    // N (deg_out then isqrt)
    float* in_isqrt  = out_isqrt + N;                // N (deg_in then isqrt)

    const int B256 = 256;
    const int featTotal = N * D_FEAT;
    dim3 blk(B256);

    // 1) degrees -> inverse sqrt
    gcn_zero_f32<<<(2 * N + B256 - 1) / B256, blk, 0, stream>>>(out_isqrt, 2 * N);
    gcn_degree<<<(E + B256 - 1) / B256, blk, 0, stream>>>(src, dst, out_isqrt, in_isqrt, E);
    gcn_deg_isqrt<<<(2 * N + B256 - 1) / B256, blk, 0, stream>>>(out_isqrt, 2 * N);

    // 2) layer 1: scale -> scatter -> transform (fuses next layer's out scale)
    gcn_scale_rows<<<(N * 16 + B256 - 1) / B256, blk, 0, stream>>>(x, out_isqrt, bufA, N * 16);
    gcn_zero_f32<<<(featTotal + B256 - 1) / B256, blk, 0, stream>>>(bufB, featTotal);
    gcn_scatter<<<((size_t)E * 16 + B256 - 1) / B256, blk, 0, stream>>>(bufA, src, dst, bufB, E);
    gcn_transform_wmma<<<(N + 15) / 16, dim3(128), 0, stream>>>(
        bufB, W1, b1, in_isqrt, out_isqrt, bufA, N);

    // 3) layer 2: scatter -> transform
    gcn_zero_f32<<<(featTotal + B256 - 1) / B256, blk, 0, stream>>>(bufB, featTotal);
    gcn_scatter<<<((size_t)E * 16 + B256 - 1) / B256, blk, 0, stream>>>(bufA, src, dst, bufB, E);
    gcn_transform_wmma<<<(N + 15) / 16, dim3(128), 0, stream>>>(
        bufB, W2, b2, in_isqrt, nullptr, bufA, N);

    // 4) classifier head
    gcn_classify<<<(N + B256 - 1) / B256, blk, 0, stream>>>(bufA, Wc, bc, out, N);
}